// Texture_41120016892628
// MI455X (gfx1250) — compile-verified
//
#include <hip/hip_runtime.h>
#include <hip/hip_bf16.h>
#include <stdint.h>

// Problem constants (from reference): T=16 textures, F=16 features, D=512,
// N=4, L=4, H=W=512.
#define TT 16
#define FFEAT 16
#define DDIM 512
#define PLANE (512 * 512)          // D*D == H*W
#define NSAMP (4 * 4 * 512 * 512)  // N*L*H*W

typedef unsigned int u32x4 __attribute__((ext_vector_type(4)));
typedef int i32x8 __attribute__((ext_vector_type(8)));
typedef int i32x4 __attribute__((ext_vector_type(4)));

// -----------------------------------------------------------------------------
// Kernel 1: TDM-based layout transpose  (T,F,D,D) -> (T,D,D,F)
// One 64(x) x 16(f) tile per wave. The Tensor Data Mover pulls 16 rows of
// 256B with a 1MB stride (tensor_dim0_stride = D*D elements) into LDS,
// inserting 2 DWORDs of padding every 64 DWORDs (row stride 66 floats) so the
// transposed LDS reads below are bank-conflict free. Output stores are fully
// coalesced 16B (float4 = 4 features) writes.
// -----------------------------------------------------------------------------
__global__ __launch_bounds__(256) void tex_transpose_tdm(const float* __restrict__ src,
                                                         float* __restrict__ dst) {
  __shared__ float sbuf[8 * 16 * 66];  // 8 waves * padded 16x66 tile
  const int lane = threadIdx.x & 31;
#if defined(__HIP_DEVICE_COMPILE__) && defined(__gfx1250__)
  const int wid = __builtin_amdgcn_readfirstlane((int)(threadIdx.x >> 5));
#else
  const int wid = (int)(threadIdx.x >> 5);
#endif
  const int tile = blockIdx.x * 8 + wid;   // 16*512*8 = 65536 tiles total
  const int t    = tile >> 12;             // / (512*8)
  const int rem  = tile & 4095;
  const int y    = rem >> 3;
  const int x0   = (rem & 7) << 6;
  // element offset of (t, f=0, y, x0) in src laid out (T,F,D,D)
  const size_t src_elem = (size_t)t * (16 * (size_t)PLANE) + (size_t)y * 512 + (size_t)x0;

#if defined(__HIP_DEVICE_COMPILE__) && defined(__gfx1250__)
  // ---- Build Tensor DMA Descriptor (D#) per CDNA5 ISA
  unsigned long long ga = (unsigned long long)(uintptr_t)(src + src_elem);
  unsigned lds_addr = (unsigned)(uintptr_t)(&sbuf[0]) + (unsigned)wid * (16 * 66 * 4);

  u32x4 g0;
  g0.x = 1u;                                        // count=1, no gather, user mode
  g0.y = lds_addr;                                  // lds_addr [63:32]
  g0.z = (unsigned)(ga & 0xFFFFFFFFull);            // global_addr[31:0]
  g0.w = (unsigned)((ga >> 32) & 0x1FFFFFFull)      // global_addr[56:32]
         | (2u << 30);                              // type=2 ("image")

  i32x8 g1;
  g1[0] = (2 << 16)      // data_size = 4 bytes
        | (1 << 20)      // pad_enable
        | (5 << 22)      // pad_interval: 64 DWORDs (one 256B row)
        | (1 << 25);     // pad_amount: 2 DWORDs  -> LDS row stride 66 floats
  g1[1] = (64 << 16);    // tensor_dim0 = 64 (lo16)
  g1[2] = (16 << 16);    // tensor_dim0 hi = 0 ; tensor_dim1 = 16 (lo16)
  g1[3] = (64 << 16);    // tensor_dim1 hi = 0 ; tile_dim0 = 64
  g1[4] = 16;            // tile_dim1 = 16 ; tile_dim2 = 0 (2D)
  g1[5] = PLANE;         // tensor_dim0_stride[31:0] = D*D elements (f-plane stride)
  g1[6] = 0;             // stride hi / tensor_dim1_stride lo (unused, 2D)
  g1[7] = 0;
  i32x4 g2 = {0, 0, 0, 0};   // D# group 2: unused (2D tensor)
  i32x4 g3 = {0, 0, 0, 0};   // D# group 3: unused (2D tensor)
  i32x8 g4 = {0, 0, 0, 0, 0, 0, 0, 0};  // extra group required by 6-arg builtin

  __builtin_amdgcn_tensor_load_to_lds(g0, g1, g2, g3, g4, 0);
  __builtin_amdgcn_s_wait_tensorcnt(0);
  asm volatile("" ::: "memory");  // TDM wrote LDS behind the compiler's back
#endif

  // ---- Transposed write-out: contiguous 1024 floats per wave tile.
  const float* wsl = &sbuf[wid * (16 * 66)];
  float4* d4 = (float4*)dst;
  const size_t ob4 = ((size_t)t * PLANE + (size_t)y * 512 + (size_t)x0) * 4;  // float4 units
  #pragma unroll
  for (int k = 0; k < 8; ++k) {
    int m4 = k * 32 + lane;       // float4 index within tile [0,256)
    int xl = m4 >> 2;             // x within tile
    int fb = (m4 & 3) << 2;       // first feature of this float4
    float4 r;
    r.x = wsl[(fb + 0) * 66 + xl];
    r.y = wsl[(fb + 1) * 66 + xl];
    r.z = wsl[(fb + 2) * 66 + xl];
    r.w = wsl[(fb + 3) * 66 + xl];
    d4[ob4 + m4] = r;
  }
}

// -----------------------------------------------------------------------------
// Kernel 2 (fast path): bilinear sampler over transposed (T,D,D,F) atlas.
// 1 thread = 1 (n,l,h,w) sample; lanes consecutive in w => coalesced uv/mask
// loads and coalesced per-feature-plane stores. Corner gathers are contiguous
// 64B (4x float4) from the L2-resident atlas; output stores are non-temporal
// so the 256MB stream doesn't evict the atlas from the 192MB L2.
// -----------------------------------------------------------------------------
__global__ __launch_bounds__(256) void sample_fast(const float* __restrict__ uv,
                                                   const int* __restrict__ mask,
                                                   const float* __restrict__ tex,
                                                   float* __restrict__ out) {
  const int i = blockIdx.x * 256 + threadIdx.x;
  const int w = i & 511;
  const int h = (i >> 9) & 511;
  const int l = (i >> 18) & 3;
  const int n = i >> 20;
  const int p = (h << 9) | w;

  const float u = __builtin_nontemporal_load(uv + (size_t)(n * 8 + 2 * l) * PLANE + p);
  const float v = __builtin_nontemporal_load(uv + (size_t)(n * 8 + 2 * l + 1) * PLANE + p);
  const int tid = __builtin_nontemporal_load(mask + (size_t)(n * 4 + l) * PLANE + p);

  // ix = clip((u+1)*256 - 0.5, 0, 511)
  const float ix = fminf(fmaxf(fmaf(u, 256.0f, 255.5f), 0.0f), 511.0f);
  const float iy = fminf(fmaxf(fmaf(v, 256.0f, 255.5f), 0.0f), 511.0f);
  const float x0f = floorf(ix), y0f = floorf(iy);
  const float wx = ix - x0f, wy = iy - y0f;
  const int x0 = (int)x0f, y0 = (int)y0f;
  const int x1 = min(x0 + 1, 511), y1 = min(y0 + 1, 511);

  const float vd = (tid >= 0 && tid < TT) ? 1.0f : 0.0f;
  const int tc = min(max(tid, 0), TT - 1);

  const float w11 = (wx * wy) * vd;
  const float w10 = wx * vd - w11;          // wx*(1-wy)*vd
  const float w01 = wy * vd - w11;          // (1-wx)*wy*vd
  const float w00 = vd - wx * vd - wy * vd + w11;  // (1-wx)*(1-wy)*vd

  const float4* t4 = (const float4*)tex;
  const size_t tb = (size_t)tc * PLANE;
  const size_t o00 = (tb + (size_t)y0 * 512 + x0) * 4;  // float4 index of feature chunk 0
  const size_t o10 = (tb + (size_t)y0 * 512 + x1) * 4;
  const size_t o01 = (tb + (size_t)y1 * 512 + x0) * 4;
  const size_t o11 = (tb + (size_t)y1 * 512 + x1) * 4;

  const size_t ob = (size_t)(n * 64 + l * 16) * PLANE + p;
  #pragma unroll
  for (int c = 0; c < 4; ++c) {
    float4 a = t4[o00 + c];
    float4 b = t4[o10 + c];
    float4 cc = t4[o01 + c];
    float4 d = t4[o11 + c];
    float rx = fmaf(d.x, w11, fmaf(cc.x, w01, fmaf(b.x, w10, a.x * w00)));
    float ry = fmaf(d.y, w11, fmaf(cc.y, w01, fmaf(b.y, w10, a.y * w00)));
    float rz = fmaf(d.z, w11, fmaf(cc.z, w01, fmaf(b.z, w10, a.z * w00)));
    float rw = fmaf(d.w, w11, fmaf(cc.w, w01, fmaf(b.w, w10, a.w * w00)));
    __builtin_nontemporal_store(rx, out + ob + (size_t)(4 * c + 0) * PLANE);
    __builtin_nontemporal_store(ry, out + ob + (size_t)(4 * c + 1) * PLANE);
    __builtin_nontemporal_store(rz, out + ob + (size_t)(4 * c + 2) * PLANE);
    __builtin_nontemporal_store(rw, out + ob + (size_t)(4 * c + 3) * PLANE);
  }
}

// -----------------------------------------------------------------------------
// Fallback (if ws_size < 256MB): sample directly from (T,F,D,D); 16 strided
// scalar loads per corner. Correct but ~16x more gather traffic.
// -----------------------------------------------------------------------------
__global__ __launch_bounds__(256) void sample_direct(const float* __restrict__ uv,
                                                     const int* __restrict__ mask,
                                                     const float* __restrict__ data,
                                                     float* __restrict__ out) {
  const int i = blockIdx.x * 256 + threadIdx.x;
  const int w = i & 511;
  const int h = (i >> 9) & 511;
  const int l = (i >> 18) & 3;
  const int n = i >> 20;
  const int p = (h << 9) | w;

  const float u = uv[(size_t)(n * 8 + 2 * l) * PLANE + p];
  const float v = uv[(size_t)(n * 8 + 2 * l + 1) * PLANE + p];
  const int tid = mask[(size_t)(n * 4 + l) * PLANE + p];

  const float ix = fminf(fmaxf(fmaf(u, 256.0f, 255.5f), 0.0f), 511.0f);
  const float iy = fminf(fmaxf(fmaf(v, 256.0f, 255.5f), 0.0f), 511.0f);
  const float x0f = floorf(ix), y0f = floorf(iy);
  const float wx = ix - x0f, wy = iy - y0f;
  const int x0 = (int)x0f, y0 = (int)y0f;
  const int x1 = min(x0 + 1, 511), y1 = min(y0 + 1, 511);
  const float vd = (tid >= 0 && tid < TT) ? 1.0f : 0.0f;
  const int tc = min(max(tid, 0), TT - 1);

  const float w11 = (wx * wy) * vd;
  const float w10 = wx * vd - w11;
  const float w01 = wy * vd - w11;
  const float w00 = vd - wx * vd - wy * vd + w11;

  float acc[16];
  #pragma unroll
  for (int f = 0; f < 16; ++f) acc[f] = 0.0f;

  const float* tbase = data + (size_t)tc * 16 * PLANE;
  {
    const float* p00 = tbase + (size_t)y0 * 512 + x0;
    const float* p10 = tbase + (size_t)y0 * 512 + x1;
    const float* p01 = tbase + (size_t)y1 * 512 + x0;
    const float* p11 = tbase + (size_t)y1 * 512 + x1;
    #pragma unroll
    for (int f = 0; f < 16; ++f) {
      const size_t fo = (size_t)f * PLANE;
      acc[f] = fmaf(p00[fo], w00, acc[f]);
      acc[f] = fmaf(p10[fo], w10, acc[f]);
      acc[f] = fmaf(p01[fo], w01, acc[f]);
      acc[f] = fmaf(p11[fo], w11, acc[f]);
    }
  }

  const size_t ob = (size_t)(n * 64 + l * 16) * PLANE + p;
  #pragma unroll
  for (int f = 0; f < 16; ++f)
    __builtin_nontemporal_store(acc[f], out + ob + (size_t)f * PLANE);
}

extern "C" void kernel_launch(void* const* d_in, const int* in_sizes, int n_in,
                              void* d_out, int out_size, void* d_ws, size_t ws_size,
                              hipStream_t stream) {
  const float* uv   = (const float*)d_in[0];  // (N, 2L, H, W) f32
  const int*   mask = (const int*)d_in[1];    // (N, L, H, W) i32
  const float* data = (const float*)d_in[2];  // (T, F, D, D) f32
  float* out = (float*)d_out;                 // (N, L*F, H, W) f32

  const size_t need = (size_t)TT * DDIM * DDIM * FFEAT * sizeof(float);  // 256 MB
  if (d_ws != nullptr && ws_size >= need) {
    float* tex = (float*)d_ws;
    // 16*512*8 = 65536 tiles, 8 waves (tiles) per 256-thread block
    tex_transpose_tdm<<<dim3(8192), dim3(256), 0, stream>>>(data, tex);
    sample_fast<<<dim3(NSAMP / 256), dim3(256), 0, stream>>>(uv, mask, tex, out);
  } else {
    sample_direct<<<dim3(NSAMP / 256), dim3(256), 0, stream>>>(uv, mask, data, out);
  }
}